// GCN_12489764896775
// MI455X (gfx1250) — compile-verified
//
#include <hip/hip_runtime.h>
#include <hip/hip_bf16.h>

// ---------------------------------------------------------------------------
// GCN forward on gfx1250 (MI455X):
//   h1 = relu(Agg(x @ W1) + b1);  h2 = relu(Agg(h1 @ W2) + b2)
//   out = mean-pool-by-graph(h2) @ Wfc + bfc
// GEMMs: native fp32 WMMA (v_wmma_f32_16x16x4_f32, wave32), weight matrix
// staged swizzled in LDS (<=256KB of the 320KB WGP LDS), persistent blocks,
// branchless register double-buffered fragments. LDS fill uses the gfx1250
// async global->LDS path (ASYNCcnt) when the toolchain exposes it.
// Edge aggregation = gather + scale + global f32 atomic scatter (L2-resident).
// ---------------------------------------------------------------------------

#define N_NODES  50000
#define N_EDGES  800000
#define N_GRAPHS 128
#define IN_C     128
#define HID_C    256
#define OUT_C    128

typedef __attribute__((ext_vector_type(2))) float v2f;
typedef __attribute__((ext_vector_type(8))) float v8f;

#if defined(__has_builtin)
#if __has_builtin(__builtin_amdgcn_global_load_async_to_lds_b32) && \
    __has_builtin(__builtin_amdgcn_s_wait_asynccnt)
#define GCN_HAS_ASYNC_LDS 1
#endif
#endif
#ifndef GCN_HAS_ASYNC_LDS
#define GCN_HAS_ASYNC_LDS 0
#endif

// ---------------------------------------------------------------------------
// fp32 WMMA GEMM: C[M,N] = A[M,K] @ B[K,N] (+bias) (optional relu), row-major.
// M%16==0, N%64==0 (and N/64 divides 8), K%4==0, K*N*4 bytes dynamic LDS.
//
// Fragment layouts (ISA 7.12.2, 32-bit data):
//   A 16x4 : lane -> M = lane%16 ; VGPR v, half h=lane/16 -> K = 2h + v
//   B 4x16 : lane -> N = lane%16 ; VGPR v, half h        -> K = 2h + v
//   C 16x16: lane -> N = lane%16 ; VGPR v, half h        -> M = v + 8h
//
// LDS B swizzle: sB[kg*4N + n*4 + kk] = B[4*kg+kk][n]  (kg = k/4, kk = 0..3)
//   -> lane fragment (kk = 2h, 2h+1) is one contiguous 8-byte ds_load.
// ---------------------------------------------------------------------------
__global__ void __launch_bounds__(256)
gcn_gemm_f32_wmma(const float* __restrict__ A, int lda,
                  const float* __restrict__ B, int ldb,
                  float* __restrict__ C, int ldc,
                  int M, int K, int N,
                  const float* __restrict__ bias, int do_relu)
{
    extern __shared__ float sB[];           // K*N floats, swizzled

#if GCN_HAS_ASYNC_LDS
    // ---- async swizzled fill: each lane DMAs one dword global->LDS with its
    // own source/dest address; retire all with one s_wait_asynccnt 0 ----
    {
        const int total = K * N;
        for (int idx = threadIdx.x; idx < total; idx += blockDim.x) {
            const int k  = idx / N;
            const int n  = idx - k * N;
            const int kg = k >> 2;
            const int kk = k & 3;
            const float* g = B + (size_t)k * ldb + n;
            float* l = sB + (size_t)kg * ((size_t)N << 2) + ((size_t)n << 2) + kk;
            __builtin_amdgcn_global_load_async_to_lds_b32((int*)g, (int*)l, 0, 0);
        }
        __builtin_amdgcn_s_wait_asynccnt(0);
    }
#else
    // ---- stage B -> LDS (coalesced global reads, ds_store_b128 writes) ----
    {
        const int kgN = (K >> 2) * N;       // number of float4 packets
        for (int idx = threadIdx.x; idx < kgN; idx += blockDim.x) {
            const int kg = idx / N;
            const int n  = idx - kg * N;
            const float* g = B + (size_t)(kg << 2) * ldb + n;
            float4 v;
            v.x = g[0];
            v.y = g[(size_t)ldb];
            v.z = g[2 * (size_t)ldb];
            v.w = g[3 * (size_t)ldb];
            *reinterpret_cast<float4*>(sB + ((size_t)idx << 2)) = v;
        }
    }
#endif
    __syncthreads();

    const int lane    = threadIdx.x & 31;
    const int waveId  = threadIdx.x >> 5;
    const int stripsN = N >> 6;                       // 64-wide column strips
    const int rowsPerBlock = (blockDim.x >> 5) / stripsN;
    const int waveRow = waveId / stripsN;
    const int waveCol = waveId - waveRow * stripsN;
    const int rowTiles = M >> 4;

    const int half = lane >> 4;                       // 0 or 1
    const int mr   = lane & 15;
    const int n0   = waveCol << 6;
    const int KG   = K >> 2;
    const int N4   = N << 2;                          // floats per k-group in sB

    // lane's base into swizzled B: + kg*N4 per k-step, + j*64 per 16-col tile
    const float* sBl = sB + (size_t)(n0 + mr) * 4 + 2 * half;

    for (int rt = blockIdx.x * rowsPerBlock + waveRow; rt < rowTiles;
         rt += gridDim.x * rowsPerBlock) {
        const int m0 = rt << 4;
        const float* Arow = A + (size_t)(m0 + mr) * lda + 2 * half;

        v8f acc[4];
        acc[0] = (v8f){0,0,0,0,0,0,0,0};
        acc[1] = acc[0]; acc[2] = acc[0]; acc[3] = acc[0];

        // branchless register double-buffer: prefetch fragment for step
        // min(kg+1, KG-1) before issuing the 4 wmmas of step kg.
        v2f a0, b0[4];
        a0.x = Arow[0];
        a0.y = Arow[1];
#pragma unroll
        for (int j = 0; j < 4; ++j) {
            b0[j].x = sBl[j * 64 + 0];
            b0[j].y = sBl[j * 64 + 1];
        }

#pragma unroll 2
        for (int kg = 0; kg < KG; ++kg) {
            const int kn = (kg + 1 < KG) ? (kg + 1) : kg;   // s_cselect, no branch
            const float* An = Arow + (kn << 2);
            v2f a1;
            a1.x = An[0];
            a1.y = An[1];
            const float* Bn = sBl + (size_t)kn * N4;
            v2f b1[4];
#pragma unroll
            for (int j = 0; j < 4; ++j) {
                b1[j].x = Bn[j * 64 + 0];
                b1[j].y = Bn[j * 64 + 1];
            }
#pragma unroll
            for (int j = 0; j < 4; ++j) {
                acc[j] = __builtin_amdgcn_wmma_f32_16x16x4_f32(
                    false, a0, false, b0[j], (short)0, acc[j], false, false);
            }
            a0 = a1;
#pragma unroll
            for (int j = 0; j < 4; ++j) b0[j] = b1[j];
        }

#pragma unroll
        for (int j = 0; j < 4; ++j) {
            const int n = n0 + j * 16 + mr;
            const float bv = bias ? bias[n] : 0.0f;
#pragma unroll
            for (int v = 0; v < 8; ++v) {
                const int m = m0 + v + 8 * half;
                float val = acc[j][v] + bv;
                if (do_relu) val = fmaxf(val, 0.0f);
                C[(size_t)m * ldc + n] = val;
            }
        }
    }
}

// ---------------------------------------------------------------------------
// Helper kernels
// ---------------------------------------------------------------------------
__global__ void gcn_zero_f32(float* __restrict__ p, int n)
{
    int i = blockIdx.x * blockDim.x + threadIdx.x;
    if (i < n) p[i] = 0.0f;
}

__global__ void gcn_fill_f32(float* __restrict__ p, float v, int n)
{
    int i = blockIdx.x * blockDim.x + threadIdx.x;
    if (i < n) p[i] = v;
}

// deg[dst[e]] += 1  (self-loop's +1 pre-seeded by gcn_fill_f32(1.0))
__global__ void gcn_deg_accum(const int* __restrict__ dst, float* __restrict__ deg,
                              int n_edges)
{
    int e = blockIdx.x * blockDim.x + threadIdx.x;
    if (e < n_edges) atomicAdd(&deg[dst[e]], 1.0f);
}

__global__ void gcn_deg_to_dinv(float* __restrict__ d, int n)
{
    int i = blockIdx.x * blockDim.x + threadIdx.x;
    if (i < n) {
        float deg = d[i];
        d[i] = (deg > 0.0f) ? rsqrtf(deg) : 0.0f;
    }
}

// out[dst, :] += dinv[src]*dinv[dst] * H[src, :]   (C = 256, 4 channels/thread)
__global__ void __launch_bounds__(256)
gcn_scatter_edges(const float* __restrict__ H, float* __restrict__ out,
                  const int* __restrict__ src, const int* __restrict__ dst,
                  const float* __restrict__ dinv, int n_edges)
{
    long long gid = (long long)blockIdx.x * blockDim.x + threadIdx.x;
    int e = (int)(gid >> 6);           // 64 chunks of 4 channels (HID_C = 256)
    int c = ((int)gid & 63) << 2;
    if (e >= n_edges) return;
    const int s = src[e], d = dst[e];
    const float nrm = dinv[s] * dinv[d];
    const float4 v = *reinterpret_cast<const float4*>(H + (size_t)s * HID_C + c);
    float* o = out + (size_t)d * HID_C + c;
    atomicAdd(o + 0, v.x * nrm);
    atomicAdd(o + 1, v.y * nrm);
    atomicAdd(o + 2, v.z * nrm);
    atomicAdd(o + 3, v.w * nrm);
}

// agg += dinv[i]^2 * H[i]  (self loop) + bias, then relu  -> in place
__global__ void gcn_finalize_conv(float* __restrict__ agg, const float* __restrict__ H,
                                  const float* __restrict__ dinv,
                                  const float* __restrict__ bias, int n_nodes)
{
    long long gid = (long long)blockIdx.x * blockDim.x + threadIdx.x;
    int i = (int)(gid >> 8);           // HID_C = 256
    int c = (int)gid & 255;
    if (i >= n_nodes) return;
    const size_t idx = (size_t)i * HID_C + c;
    const float di = dinv[i];
    float v = agg[idx] + di * di * H[idx] + bias[c];
    agg[idx] = fmaxf(v, 0.0f);
}

// sums[batch[i], :] += H[i, :]   (4 channels/thread)
__global__ void __launch_bounds__(256)
gcn_pool_accum(const float* __restrict__ H, const int* __restrict__ batch,
               float* __restrict__ sums, int n_nodes)
{
    long long gid = (long long)blockIdx.x * blockDim.x + threadIdx.x;
    int i = (int)(gid >> 6);
    int c = ((int)gid & 63) << 2;
    if (i >= n_nodes) return;
    const int g = batch[i];
    const float4 v = *reinterpret_cast<const float4*>(H + (size_t)i * HID_C + c);
    float* o = sums + (size_t)g * HID_C + c;
    atomicAdd(o + 0, v.x);
    atomicAdd(o + 1, v.y);
    atomicAdd(o + 2, v.z);
    atomicAdd(o + 3, v.w);
}

__global__ void gcn_pool_count(const int* __restrict__ batch, float* __restrict__ cnt,
                               int n_nodes)
{
    int i = blockIdx.x * blockDim.x + threadIdx.x;
    if (i < n_nodes) atomicAdd(&cnt[batch[i]], 1.0f);
}

__global__ void gcn_pool_div(float* __restrict__ sums, const float* __restrict__ cnt)
{
    int idx = blockIdx.x * blockDim.x + threadIdx.x;   // N_GRAPHS * HID_C threads
    int g = idx >> 8;                                  // HID_C = 256
    sums[idx] *= 1.0f / fmaxf(cnt[g], 1.0f);
}

// ---------------------------------------------------------------------------
// Launch
// ---------------------------------------------------------------------------
static inline int gemm_grid(int M, int N, int maxBlocks)
{
    const int stripsN = N / 64;
    const int rowsPerBlock = 8 / stripsN;              // 8 waves per block
    const int rowTiles = M / 16;
    int blocks = (rowTiles + rowsPerBlock - 1) / rowsPerBlock;
    return blocks < maxBlocks ? blocks : maxBlocks;
}

extern "C" void kernel_launch(void* const* d_in, const int* in_sizes, int n_in,
                              void* d_out, int out_size, void* d_ws, size_t ws_size,
                              hipStream_t stream)
{
    (void)in_sizes; (void)n_in; (void)out_size; (void)ws_size;

    const float* x    = (const float*)d_in[0];
    const int*   eidx = (const int*)  d_in[1];
    const int*   src  = eidx;                 // edge_index[0]
    const int*   dst  = eidx + N_EDGES;       // edge_index[1]
    const int*   batch= (const int*)  d_in[2];
    const float* W1   = (const float*)d_in[3];
    const float* b1   = (const float*)d_in[4];
    const float* W2   = (const float*)d_in[5];
    const float* b2   = (const float*)d_in[6];
    const float* Wfc  = (const float*)d_in[7];
    const float* bfc  = (const float*)d_in[8];
    float* out = (float*)d_out;

    // Workspace layout (floats): dinv | pooled sums | cnt | bufA | bufB
    float* w    = (float*)d_ws;
    float* dinv = w;                                   // N_NODES
    float* sums = dinv + N_NODES;                      // N_GRAPHS*HID_C = 32768
    float* cnt  = sums + N_GRAPHS * HID_C;             // N_GRAPHS
    float* bufA = cnt + N_GRAPHS;                      // N_NODES*HID_C (16B-aligned)
    float* bufB = bufA + (size_t)N_NODES * HID_C;      // N_NODES*HID_C

    const int nodeElems = N_NODES * HID_C;             // 12,800,000

    // ---- degrees -> dinv = rsqrt(1 + indeg) ----
    gcn_fill_f32<<<(N_NODES + 255) / 256, 256, 0, stream>>>(dinv, 1.0f, N_NODES);
    gcn_deg_accum<<<(N_EDGES + 255) / 256, 256, 0, stream>>>(dst, dinv, N_EDGES);
    gcn_deg_to_dinv<<<(N_NODES + 255) / 256, 256, 0, stream>>>(dinv, N_NODES);

    // ---- layer 1: bufA = x @ W1 ; bufB = relu(Agg(bufA) + b1) ----
    gcn_gemm_f32_wmma<<<gemm_grid(N_NODES, HID_C, 256), 256,
                        IN_C * HID_C * sizeof(float), stream>>>(
        x, IN_C, W1, HID_C, bufA, HID_C, N_NODES, IN_C, HID_C, nullptr, 0);
    gcn_zero_f32<<<(nodeElems + 255) / 256, 256, 0, stream>>>(bufB, nodeElems);
    {
        long long tot = (long long)N_EDGES * 64;
        gcn_scatter_edges<<<(unsigned)((tot + 255) / 256), 256, 0, stream>>>(
            bufA, bufB, src, dst, dinv, N_EDGES);
    }
    gcn_finalize_conv<<<(nodeElems + 255) / 256, 256, 0, stream>>>(
        bufB, bufA, dinv, b1, N_NODES);

    // ---- layer 2: bufA = bufB @ W2 ; bufB = relu(Agg(bufA) + b2) ----
    gcn_gemm_f32_wmma<<<gemm_grid(N_NODES, HID_C, 256), 256,
                        HID_C * HID_C * sizeof(float), stream>>>(
        bufB, HID_C, W2, HID_C, bufA, HID_C, N_NODES, HID_C, HID_C, nullptr, 0);
    gcn_zero_f32<<<(nodeElems + 255) / 256, 256, 0, stream>>>(bufB, nodeElems);
    {
        long long tot = (long long)N_EDGES * 64;
        gcn_scatter_edges<<<(unsigned)((tot + 255) / 256), 256, 0, stream>>>(
            bufA, bufB, src, dst, dinv, N_EDGES);
    }
    gcn_finalize_conv<<<(nodeElems + 255) / 256, 256, 0, stream>>>(
        bufB, bufA, dinv, b2, N_NODES);

    // ---- mean pool over graphs ----
    gcn_zero_f32<<<(N_GRAPHS * HID_C + N_GRAPHS + 255) / 256, 256, 0, stream>>>(
        sums, N_GRAPHS * HID_C + N_GRAPHS);
    {
        long long tot = (long long)N_NODES * 64;
        gcn_pool_accum<<<(unsigned)((tot + 255) / 256), 256, 0, stream>>>(
            bufB, batch, sums, N_NODES);
    }
    gcn_pool_count<<<(N_NODES + 255) / 256, 256, 0, stream>>>(batch, cnt, N_NODES);
    gcn_pool_div<<<(N_GRAPHS * HID_C) / 256, 256, 0, stream>>>(sums, cnt);

    // ---- out = pooled @ Wfc + bfc  (128x256 @ 256x128) ----
    gcn_gemm_f32_wmma<<<gemm_grid(N_GRAPHS, OUT_C, 256), 256,
                        HID_C * OUT_C * sizeof(float), stream>>>(
        sums, HID_C, Wfc, OUT_C, out, OUT_C, N_GRAPHS, HID_C, OUT_C, bfc, 0);
}